// NeuralWordAligner_1477468749961
// MI455X (gfx1250) — compile-verified
//
#include <hip/hip_runtime.h>
#include <hip/hip_bf16.h>
#include <math.h>

// CRF forward via exp-space WMMA GEMM on gfx1250 (wave32).
//   B=8192, L=48, T=49 (padded to 64).
//   f'[b,j] = m_b + log( P[b,:] @ ET[:,j] ) + emit[b,i,j],  P=exp(f-m), ET=exp(trans^T)

#define B_  8192
#define L_  48
#define T_  49
#define TP  64          // padded state dim
#define SROW 68         // LDS row stride in floats (bank-conflict-free)

typedef float v2f __attribute__((ext_vector_type(2)));
typedef float v8f __attribute__((ext_vector_type(8)));

// ---------------- Kernel 1: ET[k][j] = exp(trans[j,k]) padded to 64x64 ----------------
__global__ void build_ett(const float* __restrict__ tf, const float* __restrict__ w2,
                          const float* __restrict__ b2, float* __restrict__ ett) {
    int idx = blockIdx.x * blockDim.x + threadIdx.x;   // 0..4095
    if (idx >= TP * TP) return;
    int k = idx / TP;
    int j = idx % TP;
    float v = 0.0f;
    if (k < T_ && j < T_) {
        float acc = b2[0];
        #pragma unroll
        for (int f = 0; f < 6; ++f)
            acc += tf[(j * T_ + k) * 6 + f] * w2[f];
        v = __expf(acc);
    }
    ett[idx] = v;
}

// ---------------- Kernel 2: forward scan, 16 batch rows per wave ----------------
__global__ __launch_bounds__(128) void crf_forward(const float* __restrict__ em,
                                                   const float* __restrict__ ett,
                                                   float* __restrict__ out) {
    __shared__ float sP[4][16 * SROW];

    const int lane = threadIdx.x & 31;
    const int wave = threadIdx.x >> 5;
    const int half = lane >> 4;          // 0: lanes 0-15, 1: lanes 16-31
    const int lj   = lane & 15;
    const int b0   = (blockIdx.x * 4 + wave) * 16;
    float* P = sP[wave];

    // --- Preload B fragments of ET into registers: breg[kk][n] covers K=4*kk..+3, N=16*n..+15.
    // 32-bit B 4x16 layout (mirror of documented A 16x4): VGPR0 = {K=0 | K=2}, VGPR1 = {K=1 | K=3},
    // lanes hold N = lane%16.
    v2f breg[16][4];
    #pragma unroll
    for (int kk = 0; kk < 16; ++kk) {
        #pragma unroll
        for (int n = 0; n < 4; ++n) {
            int k0  = kk * 4 + half * 2;
            int col = n * 16 + lj;
            breg[kk][n].x = ett[k0 * TP + col];
            breg[kk][n].y = ett[(k0 + 1) * TP + col];
        }
    }

    // g[a][p] holds (in C-matrix layout) the current log forward value minus row offset Mb:
    // row M = p + 8*half, column j = a*16 + lj.
    float g[4][8];
    float Mb[8];
    #pragma unroll
    for (int p = 0; p < 8; ++p) Mb[p] = 0.0f;

    const float logT = __logf((float)T_);

    // --- step 0: g = emit(roll)[:,0,:] + log(T), padded cols -> -inf
    #pragma unroll
    for (int a = 0; a < 4; ++a) {
        int j = a * 16 + lj;
        #pragma unroll
        for (int p = 0; p < 8; ++p) {
            int row = b0 + p + 8 * half;
            if (j < T_) {
                int jj = (j == 0) ? (T_ - 1) : (j - 1);     // roll(+1) along T
                g[a][p] = em[((size_t)row * L_) * T_ + jj] + logT;
            } else {
                g[a][p] = -INFINITY;
            }
        }
    }

    for (int i = 1; i < L_; ++i) {
        // ---- tail of previous step: row max, accumulate offset, P = exp(g - mx) -> LDS
        #pragma unroll
        for (int p = 0; p < 8; ++p) {
            float t = fmaxf(fmaxf(g[0][p], g[1][p]), fmaxf(g[2][p], g[3][p]));
            t = fmaxf(t, __shfl_xor(t, 1));
            t = fmaxf(t, __shfl_xor(t, 2));
            t = fmaxf(t, __shfl_xor(t, 4));
            t = fmaxf(t, __shfl_xor(t, 8));   // max over 16-lane group == max over all j
            Mb[p] += t;
            #pragma unroll
            for (int a = 0; a < 4; ++a) {
                float pv = __expf(g[a][p] - t);                 // 0 for padded cols
                P[(p + 8 * half) * SROW + a * 16 + lj] = pv;    // conflict-free stores
            }
        }

        // ---- GEMM: S[16x64] = P[16x64] @ ET[64x64] via 16 K-steps x 4 N-tiles of WMMA f32 16x16x4
        v8f c[4];
        #pragma unroll
        for (int n = 0; n < 4; ++n) c[n] = (v8f){0.f,0.f,0.f,0.f,0.f,0.f,0.f,0.f};

        #pragma unroll
        for (int kk = 0; kk < 16; ++kk) {
            // A 16x4 layout: lane M = lane%16; VGPR0 = {K=0 | K=2}, VGPR1 = {K=1 | K=3}
            const v2f a2 = *(const v2f*)&P[lj * SROW + kk * 4 + 2 * half];
            #pragma unroll
            for (int n = 0; n < 4; ++n) {
                c[n] = __builtin_amdgcn_wmma_f32_16x16x4_f32(
                    false, a2, false, breg[kk][n], (short)0, c[n], false, false);
            }
        }

        // ---- g = log(S) + emit[:, i, roll(j)]
        #pragma unroll
        for (int a = 0; a < 4; ++a) {
            int j = a * 16 + lj;
            #pragma unroll
            for (int p = 0; p < 8; ++p) {
                if (j < T_) {
                    int jj  = (j == 0) ? (T_ - 1) : (j - 1);
                    int row = b0 + p + 8 * half;
                    g[a][p] = __logf(c[a][p]) + em[((size_t)row * L_ + i) * T_ + jj];
                } else {
                    g[a][p] = -INFINITY;
                }
            }
        }
    }

    // ---- final: alpha = Mb + mx + log(sum_j exp(g - mx))
    #pragma unroll
    for (int p = 0; p < 8; ++p) {
        float t = fmaxf(fmaxf(g[0][p], g[1][p]), fmaxf(g[2][p], g[3][p]));
        t = fmaxf(t, __shfl_xor(t, 1));
        t = fmaxf(t, __shfl_xor(t, 2));
        t = fmaxf(t, __shfl_xor(t, 4));
        t = fmaxf(t, __shfl_xor(t, 8));
        float s = __expf(g[0][p] - t) + __expf(g[1][p] - t)
                + __expf(g[2][p] - t) + __expf(g[3][p] - t);
        s += __shfl_xor(s, 1);
        s += __shfl_xor(s, 2);
        s += __shfl_xor(s, 4);
        s += __shfl_xor(s, 8);
        if (lj == 0) {
            out[b0 + p + 8 * half] = Mb[p] + t + __logf(s);
        }
    }
}

extern "C" void kernel_launch(void* const* d_in, const int* in_sizes, int n_in,
                              void* d_out, int out_size, void* d_ws, size_t ws_size,
                              hipStream_t stream) {
    const float* emissions   = (const float*)d_in[0];   // [8192, 48, 49]
    const float* trans_feats = (const float*)d_in[1];   // [49, 49, 6]
    const float* w2          = (const float*)d_in[2];   // [6]
    const float* b2          = (const float*)d_in[3];   // [1]
    float*       out         = (float*)d_out;           // [8192]
    float*       ett         = (float*)d_ws;            // 64*64 floats = 16 KB

    build_ett<<<(TP * TP + 255) / 256, 256, 0, stream>>>(trans_feats, w2, b2, ett);
    crf_forward<<<B_ / 64, 128, 0, stream>>>(emissions, ett, out);
}